// BAddH_38311108280780
// MI455X (gfx1250) — compile-verified
//
#include <hip/hip_runtime.h>

typedef __attribute__((ext_vector_type(2))) float v2f;
typedef __attribute__((ext_vector_type(4))) float v4f;
typedef __attribute__((ext_vector_type(8))) float v8f;

#define NB   1024
#define MX   2048
#define DD   128
#define TILE 16
#define ROWF  132   // padded LDS row stride in floats (conflict-free)
#define ROWF4 33    // same, in float4 units

__global__ __launch_bounds__(32) void mobius_tile_kernel(
    const float* __restrict__ Bm,   // [NB, DD]
    const float* __restrict__ xm,   // [MX, DD]
    float* __restrict__ out)        // [MX, NB, DD]
{
    __shared__ float xs[TILE * ROWF];
    __shared__ float bs[TILE * ROWF];
    __shared__ v2f   ab[TILE * TILE];
    __shared__ float nxs[TILE];
    __shared__ float nbs[TILE];

    const int lane = threadIdx.x;      // 0..31
    const int lm   = lane & 15;
    const int hf   = lane >> 4;        // 0 or 1
    const int j0   = blockIdx.x * TILE;
    const int i0   = blockIdx.y * TILE;

    // ---- Stage tiles to LDS: each tile is one contiguous 2048-float block ----
    {
        const v4f* xg  = (const v4f*)(xm + (size_t)i0 * DD);
        const v4f* bg  = (const v4f*)(Bm + (size_t)j0 * DD);
        v4f* xs4 = (v4f*)xs;
        v4f* bs4 = (v4f*)bs;
#pragma unroll
        for (int f = 0; f < 512; f += 32) {
            int idx = f + lane;
            int r = idx >> 5, c = idx & 31;
            xs4[r * ROWF4 + c] = xg[idx];
            bs4[r * ROWF4 + c] = bg[idx];
        }
    }
    __syncthreads();

    // ---- Row norms (branchless: lanes 0-15 -> |x_i|^2, lanes 16-31 -> |B_j|^2) ----
    {
        const v4f* rowp = (const v4f*)(hf ? bs : xs) + lm * ROWF4;
        float s = 0.f;
#pragma unroll
        for (int t = 0; t < 32; ++t) {
            v4f v = rowp[t];
            s = fmaf(v.x, v.x, s);
            s = fmaf(v.y, v.y, s);
            s = fmaf(v.z, v.z, s);
            s = fmaf(v.w, v.w, s);
        }
        float* np = hf ? &nbs[lm] : &nxs[lm];
        *np = s;
    }
    __syncthreads();

    // ---- xy tile = x_tile . B_tile^T via chained V_WMMA_F32_16X16X4_F32, K=128 ----
    // A frag (16x4 f32): lane -> M = lane&15, VGPR v -> K = 2*(lane>>4) + v
    // B frag (4x16 f32): lane -> N = lane&15, VGPR v -> K = 2*(lane>>4) + v
    v8f acc = {0.f, 0.f, 0.f, 0.f, 0.f, 0.f, 0.f, 0.f};
    {
        const float* arow = xs + lm * ROWF;
        const float* brow = bs + lm * ROWF;
#pragma unroll
        for (int ks = 0; ks < 32; ++ks) {
            int kb = ks * 4 + 2 * hf;                 // even -> 8B aligned
            v2f a2 = *(const v2f*)(arow + kb);
            v2f b2 = *(const v2f*)(brow + kb);
            acc = __builtin_amdgcn_wmma_f32_16x16x4_f32(
                      false, a2, false, b2, (short)0, acc, false, false);
        }
    }

    // ---- Per-pair coefficients: C layout lane->(N=lane&15), VGPR v->(M=v+8*(lane>>4)) ----
    {
        float nbv = nbs[lm];
        float cx  = 1.f - nbv;
#pragma unroll
        for (int v = 0; v < 8; ++v) {
            int   m     = v + 8 * hf;
            float nxv   = nxs[m];
            float t     = fmaf(2.f, acc[v], 1.f);     // 1 + 2*xy
            float denom = fmaf(nbv, nxv, t);          // 1 + 2*xy + nB*nx
            float inv   = 1.f / denom;
            v2f c;
            c.x = (t + nxv) * inv;                    // coef_B / denom
            c.y = cx * inv;                           // coef_x / denom
            ab[m * 16 + lm] = c;
        }
    }
    __syncthreads();

    // ---- Streaming output: 256 (i,j) rows x 128 floats; 512B NT store per wave/row ----
    {
        const v4f* xs4 = (const v4f*)xs;
        const v4f* bs4 = (const v4f*)bs;
        v4f* og = (v4f*)out;
#pragma unroll 4
        for (int r = 0; r < 256; ++r) {
            int il = r >> 4, jl = r & 15;
            v2f c  = ab[r];                            // LDS broadcast read
            v4f bv = bs4[jl * ROWF4 + lane];
            v4f xv = xs4[il * ROWF4 + lane];
            v4f o;
            o.x = fmaf(c.x, bv.x, c.y * xv.x);
            o.y = fmaf(c.x, bv.y, c.y * xv.y);
            o.z = fmaf(c.x, bv.z, c.y * xv.z);
            o.w = fmaf(c.x, bv.w, c.y * xv.w);
            size_t off = ((size_t)(i0 + il) * NB + (j0 + jl)) * (DD / 4) + lane;
            __builtin_nontemporal_store(o, og + off);
        }
    }
}

extern "C" void kernel_launch(void* const* d_in, const int* in_sizes, int n_in,
                              void* d_out, int out_size, void* d_ws, size_t ws_size,
                              hipStream_t stream) {
    (void)in_sizes; (void)n_in; (void)d_ws; (void)ws_size; (void)out_size;
    const float* Bm = (const float*)d_in[0];   // B: [1024, 128]
    const float* xm = (const float*)d_in[1];   // x: [2048, 128]
    float* out = (float*)d_out;                // [2048, 1024, 128]

    dim3 grid(NB / TILE, MX / TILE);           // 64 x 128 = 8192 tiles
    mobius_tile_kernel<<<grid, 32, 0, stream>>>(Bm, xm, out);
}